// NCC_fp16_730144441171
// MI455X (gfx1250) — compile-verified
//
#include <hip/hip_runtime.h>
#include <hip/hip_bf16.h>
#include <stdint.h>

// NCC (squared, win=9) over [1,1,160,192,224] f32 volumes.
// 3 separable box-filter passes (W, H, D), 5 channels (x, y, x^2, y^2, x*y).
// Pass W: async-DMA (global_load_async_to_lds_b32) staged line filter -> 5 fields in d_ws.
// Pass H: in-place 9-tap filter along H; group-of-9 unroll => static ring slots, no shifts.
// Pass D: same along D, fused with NCC + v_wmma_f32_16x16x4_f32 accumulation.
// Final: deterministic single-block reduce (no float atomics -> bitwise deterministic).
//
// Scratch: 5 * 6881280 * 4 B = 137,625,600 B fields + 168 floats partials.

#define DD 160
#define HH 192
#define WW 224
#define NVOX (DD * HH * WW)
#define WIN 9
#define PAD 4
#define INV9 (1.0f / 9.0f)
#define NCC_EPS 1e-5f

typedef float v2f __attribute__((ext_vector_type(2)));
typedef float v8f __attribute__((ext_vector_type(8)));

// ---------------- Pass 1: filter along W, build 5 product channels ----------
__global__ void ncc_pass_w(const float* __restrict__ xt, const float* __restrict__ yp,
                           float* __restrict__ f0, float* __restrict__ f1,
                           float* __restrict__ f2, float* __restrict__ f3,
                           float* __restrict__ f4) {
    __shared__ float lx[WW + 2 * PAD];
    __shared__ float ly[WW + 2 * PAD];
    const int line = blockIdx.x;                 // line = d*HH + h
    const long base = (long)line * WW;
    const int tid = threadIdx.x;

    // Zero the halo slots with plain LDS stores.
    if (tid < PAD)                           { lx[tid] = 0.0f; ly[tid] = 0.0f; }
    if (tid >= WW + PAD && tid < WW + 2*PAD) { lx[tid] = 0.0f; ly[tid] = 0.0f; }

    // Stage the interior via the CDNA5 async global->LDS DMA path (ASYNCcnt).
    // Flat->LDS addressing uses addr[31:0] (ISA 10.2), so the truncated shared
    // pointer is exactly the LDS byte offset the instruction consumes.
    if (tid < WW) {
        unsigned loffx = (unsigned)(uintptr_t)&lx[PAD + tid];
        unsigned loffy = (unsigned)(uintptr_t)&ly[PAD + tid];
        unsigned long long gx = (unsigned long long)(uintptr_t)(xt + base + tid);
        unsigned long long gy = (unsigned long long)(uintptr_t)(yp + base + tid);
        asm volatile("global_load_async_to_lds_b32 %0, %1, off"
                     :: "v"(loffx), "v"(gx) : "memory");
        asm volatile("global_load_async_to_lds_b32 %0, %1, off"
                     :: "v"(loffy), "v"(gy) : "memory");
    }
    // Each wave drains its own async DMAs before signaling the barrier; the
    // barrier then makes every wave's LDS data visible to the whole block.
    asm volatile("s_wait_asynccnt 0" ::: "memory");
    __syncthreads();

    if (tid < WW) {
        float sx = 0.f, sy = 0.f, sxx = 0.f, syy = 0.f, sxy = 0.f;
#pragma unroll
        for (int k = 0; k < WIN; ++k) {
            float xv = lx[tid + k];
            float yv = ly[tid + k];
            sx += xv;
            sy += yv;
            sxx = fmaf(xv, xv, sxx);
            syy = fmaf(yv, yv, syy);
            sxy = fmaf(xv, yv, sxy);
        }
        long idx = base + tid;
        f0[idx] = sx * INV9;
        f1[idx] = sy * INV9;
        f2[idx] = sxx * INV9;
        f3[idx] = syy * INV9;
        f4[idx] = sxy * INV9;
    }
}

// ---------------- Pass 2: filter along H, in place, 5 channels --------------
// Group-of-9 unroll: iteration h statically replaces ring slot (h mod 9); the
// 9-tap sum is order-independent, so no register shifting is needed.
__global__ void ncc_pass_h(float* __restrict__ f0, float* __restrict__ f1,
                           float* __restrict__ f2, float* __restrict__ f3,
                           float* __restrict__ f4) {
    const int g = blockIdx.x * blockDim.x + threadIdx.x;  // over D*W, exact (140*256)
    const int d = g / WW;
    const int w = g % WW;
    const long colBase = (long)d * HH * WW + w;           // + h*WW
    float r0[WIN], r1[WIN], r2[WIN], r3[WIN], r4[WIN];

#pragma unroll
    for (int j = 0; j < WIN; ++j) {     // slot j holds position j-4
        int h = j - PAD;
        float a = 0.f, b = 0.f, c = 0.f, e = 0.f, f = 0.f;
        if (h >= 0) {
            long idx = colBase + (long)h * WW;
            a = f0[idx]; b = f1[idx]; c = f2[idx]; e = f3[idx]; f = f4[idx];
        }
        r0[j] = a; r1[j] = b; r2[j] = c; r3[j] = e; r4[j] = f;
    }

    auto step = [&](int h, int u) {     // u == h % 9 (static after unroll)
        float s0 = 0.f, s1 = 0.f, s2 = 0.f, s3 = 0.f, s4 = 0.f;
#pragma unroll
        for (int k = 0; k < WIN; ++k) {
            s0 += r0[k]; s1 += r1[k]; s2 += r2[k]; s3 += r3[k]; s4 += r4[k];
        }
        long oidx = colBase + (long)h * WW;
        f0[oidx] = s0 * INV9;           // in-place safe: positions > h still unread
        f1[oidx] = s1 * INV9;
        f2[oidx] = s2 * INV9;
        f3[oidx] = s3 * INV9;
        f4[oidx] = s4 * INV9;

        int hn = h + PAD + 1;
        float a = 0.f, b = 0.f, c = 0.f, e = 0.f, f = 0.f;
        if (hn < HH) {
            long idx = colBase + (long)hn * WW;
            a = f0[idx]; b = f1[idx]; c = f2[idx]; e = f3[idx]; f = f4[idx];
        }
        r0[u] = a; r1[u] = b; r2[u] = c; r3[u] = e; r4[u] = f;
    };

    constexpr int FULLH = (HH / WIN) * WIN;   // 189, and 189 % 9 == 0
    for (int h0 = 0; h0 < FULLH; h0 += WIN) {
#pragma unroll
        for (int u = 0; u < WIN; ++u) step(h0 + u, u);
    }
#pragma unroll
    for (int u = 0; u < HH - FULLH; ++u) step(FULLH + u, u);   // 3-iter tail
}

// -------- Pass 3: filter along D + NCC + WMMA-accumulated block reduce ------
__global__ void ncc_pass_d_reduce(const float* __restrict__ f0, const float* __restrict__ f1,
                                  const float* __restrict__ f2, const float* __restrict__ f3,
                                  const float* __restrict__ f4, float* __restrict__ partial) {
    const int g = blockIdx.x * blockDim.x + threadIdx.x;  // over H*W, exact (168*256)
    const int h = g / WW;
    const int w = g % WW;
    const long colBase = (long)h * WW + w;                // + d*HH*WW
    const long dStride = (long)HH * WW;

    float r0[WIN], r1[WIN], r2[WIN], r3[WIN], r4[WIN];
#pragma unroll
    for (int j = 0; j < WIN; ++j) {
        int d = j - PAD;
        float a = 0.f, b = 0.f, c = 0.f, e = 0.f, f = 0.f;
        if (d >= 0) {
            long idx = colBase + (long)d * dStride;
            a = f0[idx]; b = f1[idx]; c = f2[idx]; e = f3[idx]; f = f4[idx];
        }
        r0[j] = a; r1[j] = b; r2[j] = c; r3[j] = e; r4[j] = f;
    }

    // WMMA accumulator: D += ones(16x4) x B(4x16) => sum(D) == 16 * sum(B),
    // independent of the exact lane<->matrix mapping (A is all ones, the
    // (lane,element)->entry maps are bijections). EXEC is all-1s: exact grid,
    // no divergent control flow around the WMMA.
    v8f acc = {};
    const v2f ones = {1.0f, 1.0f};

    auto stepd = [&](int d, int u) {
        float s0 = 0.f, s1 = 0.f, s2 = 0.f, s3 = 0.f, s4 = 0.f;
#pragma unroll
        for (int k = 0; k < WIN; ++k) {
            s0 += r0[k]; s1 += r1[k]; s2 += r2[k]; s3 += r3[k]; s4 += r4[k];
        }
        float mux = s0 * INV9, muy = s1 * INV9;
        float ex2 = s2 * INV9, ey2 = s3 * INV9, exy = s4 * INV9;
        float sigx = ex2 - mux * mux;
        float sigy = ey2 - muy * muy;
        float sigxy = exy - mux * muy;
        float den = sigx * sigy + NCC_EPS;
        // v_rcp_f32 (~1 ulp) instead of the IEEE div expansion: the hot loop
        // must stay bandwidth-bound, not VALU-bound.
        float ncc = (sigxy * sigxy) * __builtin_amdgcn_rcpf(den);

        v2f b;
        b.x = ncc;
        b.y = 0.0f;
        acc = __builtin_amdgcn_wmma_f32_16x16x4_f32(
            false, ones, false, b, (short)0, acc, false, false);

        int dn = d + PAD + 1;
        float a = 0.f, b2 = 0.f, c = 0.f, e = 0.f, f = 0.f;
        if (dn < DD) {
            long idx = colBase + (long)dn * dStride;
            a = f0[idx]; b2 = f1[idx]; c = f2[idx]; e = f3[idx]; f = f4[idx];
        }
        r0[u] = a; r1[u] = b2; r2[u] = c; r3[u] = e; r4[u] = f;
    };

    constexpr int FULLD = (DD / WIN) * WIN;   // 153, and 153 % 9 == 0
    for (int d0 = 0; d0 < FULLD; d0 += WIN) {
#pragma unroll
        for (int u = 0; u < WIN; ++u) stepd(d0 + u, u);
    }
#pragma unroll
    for (int u = 0; u < DD - FULLD; ++u) stepd(FULLD + u, u);  // 7-iter tail

    // Per-lane partial of the distributed D accumulator, then deterministic LDS tree.
    float s = acc[0] + acc[1] + acc[2] + acc[3] + acc[4] + acc[5] + acc[6] + acc[7];
    __shared__ float red[256];
    const int tid = threadIdx.x;
    red[tid] = s;
    __syncthreads();
#pragma unroll
    for (int off = 128; off > 0; off >>= 1) {
        if (tid < off) red[tid] += red[tid + off];
        __syncthreads();
    }
    if (tid == 0) partial[blockIdx.x] = red[0];
}

// ---------------- Final: deterministic single-block reduce ------------------
__global__ void ncc_final(const float* __restrict__ partial, int n, float* __restrict__ out) {
    __shared__ float red[256];
    const int tid = threadIdx.x;
    float s = (tid < n) ? partial[tid] : 0.0f;
    red[tid] = s;
    __syncthreads();
#pragma unroll
    for (int off = 128; off > 0; off >>= 1) {
        if (tid < off) red[tid] += red[tid + off];
        __syncthreads();
    }
    if (tid == 0) {
        // acc carries a x16 factor from A = ones(16x4); negate + mean.
        out[0] = -(red[0] * (1.0f / 16.0f)) / (float)NVOX;
    }
}

extern "C" void kernel_launch(void* const* d_in, const int* in_sizes, int n_in,
                              void* d_out, int out_size, void* d_ws, size_t ws_size,
                              hipStream_t stream) {
    const float* y_true = (const float*)d_in[0];
    const float* y_pred = (const float*)d_in[1];
    float* ws = (float*)d_ws;

    float* f0 = ws + 0L * NVOX;
    float* f1 = ws + 1L * NVOX;
    float* f2 = ws + 2L * NVOX;
    float* f3 = ws + 3L * NVOX;
    float* f4 = ws + 4L * NVOX;
    float* partial = ws + 5L * NVOX;   // 168 floats

    const int nBlocksH = (DD * WW) / 256;   // 140, exact
    const int nBlocksD = (HH * WW) / 256;   // 168, exact

    ncc_pass_w<<<DD * HH, 256, 0, stream>>>(y_true, y_pred, f0, f1, f2, f3, f4);
    ncc_pass_h<<<nBlocksH, 256, 0, stream>>>(f0, f1, f2, f3, f4);
    ncc_pass_d_reduce<<<nBlocksD, 256, 0, stream>>>(f0, f1, f2, f3, f4, partial);
    ncc_final<<<1, 256, 0, stream>>>(partial, nBlocksD, (float*)d_out);
}